// EnvironmentAwareRouter_59631325937724
// MI455X (gfx1250) — compile-verified
//
#include <hip/hip_runtime.h>
#include <math.h>

typedef float v2f __attribute__((ext_vector_type(2)));
typedef float v8f __attribute__((ext_vector_type(8)));

// gfx1250 has hardware V_TANH_F32 (TRANS op) -- branch-free, co-executes with VALU.
#if defined(__has_builtin)
#if __has_builtin(__builtin_amdgcn_tanhf)
#define FAST_TANH(x) __builtin_amdgcn_tanhf(x)
#elif __has_builtin(__builtin_amdgcn_tanh_f32)
#define FAST_TANH(x) __builtin_amdgcn_tanh_f32(x)
#endif
#endif
#ifndef FAST_TANH
#define FAST_TANH(x) tanhf(x)
#endif

__device__ __forceinline__ float gelu_fast(float x) {
    // tanh-form gelu: ~5 VALU + 1 TRANS (vs ~30 branchy VALU for erff); 109M evals
    float x3 = x * x * x;
    float t  = FAST_TANH(0.7978845608f * (x + 0.044715f * x3));
    return 0.5f * x * (1.0f + t);
}
__device__ __forceinline__ float gelu_exact(float x) {
    return 0.5f * x * (1.0f + erff(x * 0.70710678118f));
}

#define SWZ_ADD(p, xm) \
    p += __int_as_float(__builtin_amdgcn_ds_swizzle(__float_as_int(p), ((xm) << 10) | 0x1f))

// B=131072, C=13, T=24, H=64, E=8, K(top)=3.  128 batch per 256-thread block.
__global__ __launch_bounds__(256, 2)
void EnvironmentAwareRouter_59631325937724_kernel(
    const float* __restrict__ ctx,   // (B,13,24)
    const float* __restrict__ gum,   // (B,8)
    const float* __restrict__ w1,    // (24,64)
    const float* __restrict__ b1,    // (64)
    const float* __restrict__ w2,    // (64,1)
    const float* __restrict__ b2,    // (1)
    const float* __restrict__ cw1,   // (13,64)
    const float* __restrict__ cb1,   // (64)
    const float* __restrict__ cw2,   // (64,8)
    const float* __restrict__ cb2,   // (8)
    float* __restrict__ out,         // [mask (B,8) | probs (B,8)]
    int Btot)
{
    __shared__ float sx[128 * 14];   // x[b_local][c], stride 14 to dodge bank conflicts

    const int lane = threadIdx.x & 31;
    const int wave = threadIdx.x >> 5;
    const int wgB  = blockIdx.x * 128;
    const int col  = lane & 15;          // N (and M) index within 16-lane half
    const int kk   = (lane >> 4) << 1;   // K sub-offset: 0 (lanes 0-15) or 2 (lanes 16-31)

    // ---- preload WMMA B-fragments of t_fc1_w (24x64): 4 N-tiles x 6 K-steps ----
    v2f   Bf[4][6];
    float b1v[4], w2v[4];
#pragma unroll
    for (int nt = 0; nt < 4; ++nt) {
        b1v[nt] = b1[nt * 16 + col];
        w2v[nt] = w2[nt * 16 + col];
#pragma unroll
        for (int ks = 0; ks < 6; ++ks) {
            const float* wp = w1 + (ks * 4 + kk) * 64 + nt * 16 + col;
            v2f bv; bv[0] = wp[0]; bv[1] = wp[64];   // K=+0,+1 of this lane's K pair
            Bf[nt][ks] = bv;
        }
    }
    const float bias2 = b2[0];

    // ---- phase 1: temporal MLP via V_WMMA_F32_16X16X4_F32 ----
    const int bb = wgB + wave * 16;                  // this wave's 16 batch elements
    if (bb + 16 <= Btot) {
        const long rowBase = (long)bb * 13;          // 208 (b,c) rows = 13 M-tiles of 16
#pragma unroll 1
        for (int t = 0; t < 13; ++t) {
            const float* ap = ctx + (rowBase + t * 16 + col) * 24 + kk;
            if (t < 12) __builtin_prefetch(ap + 16 * 24, 0, 3);

            v8f acc[4];
#pragma unroll
            for (int nt = 0; nt < 4; ++nt) {
                v8f z;
#pragma unroll
                for (int i = 0; i < 8; ++i) z[i] = 0.0f;      // zero C (inline-0 eligible)
                acc[nt] = z;
            }
#pragma unroll
            for (int ks = 0; ks < 6; ++ks) {
                v2f a = *(const v2f*)(ap + ks * 4);           // aligned b64 load
#pragma unroll
                for (int nt = 0; nt < 4; ++nt)
                    acc[nt] = __builtin_amdgcn_wmma_f32_16x16x4_f32(
                        false, a, false, Bf[nt][ks], (short)0, acc[nt], false, false);
            }
            // epilogue: +bias, gelu, dot with w2, reduce across 16-lane halves
#pragma unroll
            for (int j = 0; j < 8; ++j) {
                float p = 0.0f;
#pragma unroll
                for (int nt = 0; nt < 4; ++nt)
                    p = fmaf(gelu_fast(acc[nt][j] + b1v[nt]), w2v[nt], p);
                SWZ_ADD(p, 1); SWZ_ADD(p, 2); SWZ_ADD(p, 4); SWZ_ADD(p, 8);
                if (col == 0) {
                    int rl = t * 16 + j + (kk << 2);          // row-in-wave (+8 for upper half)
                    int bo = rl / 13;
                    int cc = rl - bo * 13;
                    sx[(wave * 16 + bo) * 14 + cc] = p + bias2;
                }
            }
        }
    }
    __syncthreads();

    // ---- phase 2: context MLP + gumbel softmax + top-3 (one thread per batch elem) ----
    if (threadIdx.x < 128) {
        int b = wgB + (int)threadIdx.x;
        if (b < Btot) {
            float x[13];
#pragma unroll
            for (int c = 0; c < 13; ++c) x[c] = sx[threadIdx.x * 14 + c];

            float lg[8];
#pragma unroll
            for (int e = 0; e < 8; ++e) lg[e] = cb2[e];
#pragma unroll 4
            for (int h = 0; h < 64; ++h) {                    // weights uniform -> s_loads
                float a = cb1[h];
#pragma unroll
                for (int c = 0; c < 13; ++c) a = fmaf(x[c], cw1[c * 64 + h], a);
                float g = gelu_exact(a);                      // exact erf feeds softmax/top-k
#pragma unroll
                for (int e = 0; e < 8; ++e) lg[e] = fmaf(g, cw2[h * 8 + e], lg[e]);
            }

            const float* gp = gum + (long)b * 8;
            float z[8], zm = -3.0e38f;
#pragma unroll
            for (int e = 0; e < 8; ++e) { z[e] = lg[e] + gp[e]; zm = fmaxf(zm, z[e]); }
            float s = 0.0f, pb[8];
#pragma unroll
            for (int e = 0; e < 8; ++e) { pb[e] = __expf(z[e] - zm); s += pb[e]; }
            float rinv = 1.0f / s;
#pragma unroll
            for (int e = 0; e < 8; ++e) pb[e] *= rinv;

            unsigned used = 0;                                // top-3, low index wins ties
#pragma unroll
            for (int k = 0; k < 3; ++k) {
                float best = -3.0e38f; int bi = 0;
#pragma unroll
                for (int e = 0; e < 8; ++e)
                    if (!((used >> e) & 1u) && pb[e] > best) { best = pb[e]; bi = e; }
                used |= 1u << bi;
            }

            float4 m0, m1, p0, p1;
            m0.x = (used >> 0) & 1u; m0.y = (used >> 1) & 1u;
            m0.z = (used >> 2) & 1u; m0.w = (used >> 3) & 1u;
            m1.x = (used >> 4) & 1u; m1.y = (used >> 5) & 1u;
            m1.z = (used >> 6) & 1u; m1.w = (used >> 7) & 1u;
            p0.x = pb[0]; p0.y = pb[1]; p0.z = pb[2]; p0.w = pb[3];
            p1.x = pb[4]; p1.y = pb[5]; p1.z = pb[6]; p1.w = pb[7];

            float4* mo = (float4*)(out + (long)b * 8);
            float4* po = (float4*)(out + (long)Btot * 8 + (long)b * 8);
            mo[0] = m0; mo[1] = m1;
            po[0] = p0; po[1] = p1;
        }
    }
}

extern "C" void kernel_launch(void* const* d_in, const int* in_sizes, int n_in,
                              void* d_out, int out_size, void* d_ws, size_t ws_size,
                              hipStream_t stream) {
    const float* ctx = (const float*)d_in[0];
    const float* gum = (const float*)d_in[1];
    const float* w1  = (const float*)d_in[2];
    const float* b1  = (const float*)d_in[3];
    const float* w2  = (const float*)d_in[4];
    const float* b2  = (const float*)d_in[5];
    const float* cw1 = (const float*)d_in[6];
    const float* cb1 = (const float*)d_in[7];
    const float* cw2 = (const float*)d_in[8];
    const float* cb2 = (const float*)d_in[9];

    int Btot = in_sizes[1] / 8;                // gumbel_noise is (B, 8)
    int grid = (Btot + 127) / 128;

    hipLaunchKernelGGL(EnvironmentAwareRouter_59631325937724_kernel,
                       dim3(grid), dim3(256), 0, stream,
                       ctx, gum, w1, b1, w2, b2, cw1, cb1, cw2, cb2,
                       (float*)d_out, Btot);
}